// Net_71030169141498
// MI455X (gfx1250) — compile-verified
//
#include <hip/hip_runtime.h>
#include <math.h>

typedef __attribute__((ext_vector_type(2))) float v2f;
typedef __attribute__((ext_vector_type(8))) float v8f;

#define TPB 256

// ---------------------------------------------------------------------------
// 1) init: deg = 1 (self loop), zero scatter accumulators
// ---------------------------------------------------------------------------
__global__ void k_init(float* __restrict__ deg, float* __restrict__ acc1,
                       float* __restrict__ acc2, int n) {
    int i = blockIdx.x * blockDim.x + threadIdx.x;
    if (i < n) {
        deg[i] = 1.0f;
        float4 z = make_float4(0.f, 0.f, 0.f, 0.f);
        ((float4*)acc1)[i] = z;
        acc2[i * 3 + 0] = 0.f;
        acc2[i * 3 + 1] = 0.f;
        acc2[i * 3 + 2] = 0.f;
    }
}

// ---------------------------------------------------------------------------
// 2) degree scatter: deg[dst[e]] += 1
// ---------------------------------------------------------------------------
__global__ void k_deg(const int* __restrict__ dst, float* __restrict__ deg, int E) {
    int e = blockIdx.x * blockDim.x + threadIdx.x;
    if (e < E) atomicAdd(&deg[dst[e]], 1.0f);
}

// ---------------------------------------------------------------------------
// 3) dinv = rsqrt(deg)   (deg >= 1 always, self loop included)
// ---------------------------------------------------------------------------
__global__ void k_dinv(float* __restrict__ deg, int n) {
    int i = blockIdx.x * blockDim.x + threadIdx.x;
    if (i < n) deg[i] = rsqrtf(deg[i]);
}

// ---------------------------------------------------------------------------
// 4) hs1 = (x @ W1) * dinv  via V_WMMA_F32_16X16X4_F32.
//    One wave = 16 nodes. A[16x4]: cols 0-1 = x, cols 2-3 = 0.
//    A layout: lanes 0-15 hold K=0,1 (M=lane); lanes 16-31 hold K=2,3.
//    B[4x16]: rows 0-1 = W1[2x4] in cols 0-3, rest 0.
//    D[m][j]: vgpr (m&7), lane j (m<8) or 16+j (m>=8).
//    Wave-uniform full-tile fast path: straight-line dinv loads + stores.
// ---------------------------------------------------------------------------
__global__ void k_transform1_wmma(const float* __restrict__ x,
                                  const float* __restrict__ W1,
                                  const float* __restrict__ dinv,
                                  float* __restrict__ hs1, int n) {
    int wid  = (blockIdx.x * blockDim.x + threadIdx.x) >> 5;
    int lane = threadIdx.x & 31;
    int base = wid * 16;
    if (base >= n) return;                  // wave-uniform

    int  col = lane & 15;
    bool hi  = lane >= 16;
    int  rbase = base + (hi ? 8 : 0);

    v2f b = {0.f, 0.f};
    if (!hi && col < 4) {                   // W1 is [2][4]; rows K=2,3 are zero
        b.x = W1[col];
        b.y = W1[4 + col];
    }

    if (base + 16 <= n) {                   // ---- full tile (common path) ----
        v2f a = {0.f, 0.f};
        if (!hi) {
            int node = base + lane;
            a.x = x[node * 2 + 0];
            a.y = x[node * 2 + 1];
        }
        v8f c = {};
        c = __builtin_amdgcn_wmma_f32_16x16x4_f32(false, a, false, b,
                                                  (short)0, c, false, false);
        if (col < 4) {
            float dv[8];
#pragma unroll
            for (int r = 0; r < 8; ++r) dv[r] = dinv[rbase + r];
#pragma unroll
            for (int r = 0; r < 8; ++r) hs1[(rbase + r) * 4 + col] = c[r] * dv[r];
        }
    } else {                                // ---- tail tile (guarded) ----
        v2f a = {0.f, 0.f};
        if (!hi) {
            int node = base + lane;
            if (node < n) { a.x = x[node * 2 + 0]; a.y = x[node * 2 + 1]; }
        }
        v8f c = {};
        c = __builtin_amdgcn_wmma_f32_16x16x4_f32(false, a, false, b,
                                                  (short)0, c, false, false);
        if (col < 4) {
#pragma unroll
            for (int r = 0; r < 8; ++r) {
                int node = rbase + r;
                if (node < n) hs1[node * 4 + col] = c[r] * dinv[node];
            }
        }
    }
}

// ---------------------------------------------------------------------------
// 5) edge scatter, 4 features: acc1[dst] += hs1[src]  (float4 gather,
//    4 non-returning global_atomic_add_f32)
// ---------------------------------------------------------------------------
__global__ void k_scatter4(const int* __restrict__ src, const int* __restrict__ dst,
                           const float* __restrict__ hs, float* __restrict__ acc, int E) {
    int e = blockIdx.x * blockDim.x + threadIdx.x;
    if (e >= E) return;
    int s = src[e];
    int d = dst[e];
    float4 v = ((const float4*)hs)[s];      // L2-resident random gather
    atomicAdd(&acc[d * 4 + 0], v.x);
    atomicAdd(&acc[d * 4 + 1], v.y);
    atomicAdd(&acc[d * 4 + 2], v.z);
    atomicAdd(&acc[d * 4 + 3], v.w);
}

// ---------------------------------------------------------------------------
// 6) layer-1 finalize fused with layer-2 transform (WMMA, K=4 exact):
//    f  = relu((acc1 + hs1) * dinv + b1)        [16x4 tile, A operand]
//    hs2 = (f @ W2) * dinv                      [W2 is 4x3 -> B 4x16 padded]
//    Lane L supplies A rows M=L&15, K = 2*(L>>4)+{0,1}.
// ---------------------------------------------------------------------------
__global__ void k_fin1_t2_wmma(const float* __restrict__ acc1,
                               const float* __restrict__ hs1,
                               const float* __restrict__ dinv,
                               const float* __restrict__ b1,
                               const float* __restrict__ W2,
                               float* __restrict__ hs2, int n) {
    int wid  = (blockIdx.x * blockDim.x + threadIdx.x) >> 5;
    int lane = threadIdx.x & 31;
    int base = wid * 16;
    if (base >= n) return;

    int  m   = lane & 15;
    int  kb  = (lane >> 4) * 2;             // features handled by this lane: kb, kb+1
    int  col = lane & 15;
    bool hi  = lane >= 16;
    int  rbase = base + (hi ? 8 : 0);
    int  node  = base + m;

    v2f b = {0.f, 0.f};
    if (col < 3) {                          // W2 is [4][3]; B rows K=kb, kb+1
        b.x = W2[(kb + 0) * 3 + col];
        b.y = W2[(kb + 1) * 3 + col];
    }

    if (base + 16 <= n) {                   // ---- full tile (common path) ----
        float di = dinv[node];
        float f0 = (acc1[node * 4 + kb]     + hs1[node * 4 + kb])     * di + b1[kb];
        float f1 = (acc1[node * 4 + kb + 1] + hs1[node * 4 + kb + 1]) * di + b1[kb + 1];
        v2f a;
        a.x = fmaxf(f0, 0.f);
        a.y = fmaxf(f1, 0.f);
        v8f c = {};
        c = __builtin_amdgcn_wmma_f32_16x16x4_f32(false, a, false, b,
                                                  (short)0, c, false, false);
        if (col < 3) {
            float dv[8];
#pragma unroll
            for (int r = 0; r < 8; ++r) dv[r] = dinv[rbase + r];
#pragma unroll
            for (int r = 0; r < 8; ++r) hs2[(rbase + r) * 3 + col] = c[r] * dv[r];
        }
    } else {                                // ---- tail tile (guarded) ----
        v2f a = {0.f, 0.f};
        if (node < n) {
            float di = dinv[node];
            float f0 = (acc1[node * 4 + kb]     + hs1[node * 4 + kb])     * di + b1[kb];
            float f1 = (acc1[node * 4 + kb + 1] + hs1[node * 4 + kb + 1]) * di + b1[kb + 1];
            a.x = fmaxf(f0, 0.f);
            a.y = fmaxf(f1, 0.f);
        }
        v8f c = {};
        c = __builtin_amdgcn_wmma_f32_16x16x4_f32(false, a, false, b,
                                                  (short)0, c, false, false);
        if (col < 3) {
#pragma unroll
            for (int r = 0; r < 8; ++r) {
                int nd = rbase + r;
                if (nd < n) hs2[nd * 3 + col] = c[r] * dinv[nd];
            }
        }
    }
}

// ---------------------------------------------------------------------------
// 7) edge scatter, 3 features
// ---------------------------------------------------------------------------
__global__ void k_scatter3(const int* __restrict__ src, const int* __restrict__ dst,
                           const float* __restrict__ hs, float* __restrict__ acc, int E) {
    int e = blockIdx.x * blockDim.x + threadIdx.x;
    if (e >= E) return;
    int s = src[e];
    int d = dst[e];
    float v0 = hs[s * 3 + 0];
    float v1 = hs[s * 3 + 1];
    float v2 = hs[s * 3 + 2];
    atomicAdd(&acc[d * 3 + 0], v0);
    atomicAdd(&acc[d * 3 + 1], v1);
    atomicAdd(&acc[d * 3 + 2], v2);
}

// ---------------------------------------------------------------------------
// 8) layer-2 finalize (sigmoid) + MLP 3->4->3->1 (27 MACs/node, VALU)
// ---------------------------------------------------------------------------
__global__ void k_fin2_mlp(const float* __restrict__ acc2,
                           const float* __restrict__ hs2,
                           const float* __restrict__ dinv,
                           const float* __restrict__ b2,
                           const float* __restrict__ W3, const float* __restrict__ b3,
                           const float* __restrict__ W4, const float* __restrict__ b4,
                           const float* __restrict__ W5, const float* __restrict__ b5,
                           float* __restrict__ out, int n) {
    int i = blockIdx.x * blockDim.x + threadIdx.x;
    if (i >= n) return;
    float di = dinv[i];
    float g[3];
#pragma unroll
    for (int k = 0; k < 3; ++k) {
        float v = (acc2[i * 3 + k] + hs2[i * 3 + k]) * di + b2[k];
        g[k] = 1.0f / (1.0f + expf(-v));    // sigmoid
    }
    float t[4];
#pragma unroll
    for (int j = 0; j < 4; ++j) {           // W3 [3][4]
        float v = b3[j];
#pragma unroll
        for (int k = 0; k < 3; ++k) v = fmaf(g[k], W3[k * 4 + j], v);
        t[j] = fmaxf(v, 0.f);
    }
    float u[3];
#pragma unroll
    for (int k = 0; k < 3; ++k) {           // W4 [4][3]
        float v = b4[k];
#pragma unroll
        for (int j = 0; j < 4; ++j) v = fmaf(t[j], W4[j * 3 + k], v);
        u[k] = fmaxf(v, 0.f);
    }
    float o = b5[0];                        // W5 [3][1]
#pragma unroll
    for (int k = 0; k < 3; ++k) o = fmaf(u[k], W5[k], o);
    out[i] = o;
}

// ---------------------------------------------------------------------------
// launcher
// ---------------------------------------------------------------------------
extern "C" void kernel_launch(void* const* d_in, const int* in_sizes, int n_in,
                              void* d_out, int out_size, void* d_ws, size_t ws_size,
                              hipStream_t stream) {
    const float* x  = (const float*)d_in[0];
    const int*   ei = (const int*)  d_in[1];
    const float* W1 = (const float*)d_in[2];
    const float* b1 = (const float*)d_in[3];
    const float* W2 = (const float*)d_in[4];
    const float* b2 = (const float*)d_in[5];
    const float* W3 = (const float*)d_in[6];
    const float* b3 = (const float*)d_in[7];
    const float* W4 = (const float*)d_in[8];
    const float* b4 = (const float*)d_in[9];
    const float* W5 = (const float*)d_in[10];
    const float* b5 = (const float*)d_in[11];

    const int n = in_sizes[0] / 2;          // [n, 2]
    const int E = in_sizes[1] / 2;          // [2, E]
    const int* src = ei;
    const int* dst = ei + E;

    // workspace layout (all offsets 16B-aligned for the given sizes)
    char* ws = (char*)d_ws;
    float* dinv = (float*)(ws);                                   // n
    float* hs1  = (float*)(ws + (size_t)n * 4);                   // n*4
    float* acc1 = (float*)(ws + (size_t)n * 4 + (size_t)n * 16);  // n*4
    float* hs2  = (float*)(ws + (size_t)n * 4 + (size_t)n * 32);  // n*3
    float* acc2 = (float*)(ws + (size_t)n * 4 + (size_t)n * 32 + (size_t)n * 12); // n*3

    const int nodeBlocks = (n + TPB - 1) / TPB;
    const int edgeBlocks = (E + TPB - 1) / TPB;
    const int tileBlocks = (n + (TPB / 32) * 16 - 1) / ((TPB / 32) * 16); // 128 nodes/block

    k_init<<<nodeBlocks, TPB, 0, stream>>>(dinv, acc1, acc2, n);
    k_deg<<<edgeBlocks, TPB, 0, stream>>>(dst, dinv, E);
    k_dinv<<<nodeBlocks, TPB, 0, stream>>>(dinv, n);
    k_transform1_wmma<<<tileBlocks, TPB, 0, stream>>>(x, W1, dinv, hs1, n);
    k_scatter4<<<edgeBlocks, TPB, 0, stream>>>(src, dst, hs1, acc1, E);
    k_fin1_t2_wmma<<<tileBlocks, TPB, 0, stream>>>(acc1, hs1, dinv, b1, W2, hs2, n);
    k_scatter3<<<edgeBlocks, TPB, 0, stream>>>(src, dst, hs2, acc2, E);
    k_fin2_mlp<<<nodeBlocks, TPB, 0, stream>>>(acc2, hs2, dinv, b2, W3, b3, W4, b4,
                                               W5, b5, (float*)d_out, n);
}